// TreeAttention_2860448219908
// MI455X (gfx1250) — compile-verified
//
#include <hip/hip_runtime.h>
#include <hip/hip_bf16.h>

#define N_NODES 65536
#define BATCH   64
#define DIM     1024

typedef __attribute__((ext_vector_type(2))) float v2f;
typedef __attribute__((ext_vector_type(8))) float v8f;

// ---------------------------------------------------------------------------
// Kernel 1: segment offsets from int64 lengths (prefix sum over 64 entries)
// ---------------------------------------------------------------------------
__global__ __launch_bounds__(32) void seg_offsets_kernel(
    const long long* __restrict__ lens, int* __restrict__ offs) {
  if (threadIdx.x == 0) {
    int acc = 0;
    for (int b = 0; b < BATCH; ++b) { offs[b] = acc; acc += (int)lens[b]; }
    offs[BATCH] = acc;  // == N_NODES
  }
}

// ---------------------------------------------------------------------------
// Kernel 2: q[b,d] = sum_h dec[b,h] * W[h,d]   (64x1024 @ 1024x1024)
// Full f32 precision via V_WMMA_F32_16X16X4_F32. One wave per 16x16 tile.
// A (16x4 f32): lanes 0-15 hold M=lane, K={0,1}; lanes 16-31 hold K={2,3}.
// B (4x16 f32): lanes 0-15 hold N=lane, K={0,1}; lanes 16-31 hold K={2,3}.
// D (16x16 f32): VGPR r -> M=r (lanes 0-15) / M=r+8 (lanes 16-31), N=lane&15.
// ---------------------------------------------------------------------------
__global__ __launch_bounds__(32) void qgemm_wmma_kernel(
    const float* __restrict__ dec,  // [64,1024]
    const float* __restrict__ W,    // [1024,1024] row-major [h, d]
    float* __restrict__ q) {        // [64,1024]
  const int lane = threadIdx.x;
  const int n0   = blockIdx.x * 16;  // d_mem tile
  const int m0   = blockIdx.y * 16;  // batch tile
  const int half = lane >> 4;        // 0: K lo pair, 1: K hi pair
  const int l16  = lane & 15;

  v8f acc = {};
  for (int k = 0; k < DIM; k += 4) {
    const int kk = k + 2 * half;
    // A fragment: dec row (m0+l16), two consecutive K values (8B aligned)
    v2f a = *(const v2f*)(dec + (size_t)(m0 + l16) * DIM + kk);
    // B fragment: W rows kk, kk+1 at column n0+l16 (coalesced across lanes)
    const float* bp = W + (size_t)kk * DIM + n0 + l16;
    v2f b; b.x = bp[0]; b.y = bp[DIM];
    acc = __builtin_amdgcn_wmma_f32_16x16x4_f32(
        /*neg_a=*/false, a, /*neg_b=*/false, b,
        /*c_mod=*/(short)0, acc, /*reuse_a=*/false, /*reuse_b=*/false);
  }
  const int row_base = m0 + half * 8;
#pragma unroll
  for (int r = 0; r < 8; ++r)
    q[(size_t)(row_base + r) * DIM + n0 + l16] = acc[r];
}

// ---------------------------------------------------------------------------
// Kernel 3: scores[n] = <memory_bank[n], q[seg(n)]>  — one wave32 per node
// ---------------------------------------------------------------------------
__global__ __launch_bounds__(256) void scores_kernel(
    const float* __restrict__ M, const float* __restrict__ q,
    const int* __restrict__ offs, float* __restrict__ scores) {
  const int wave = threadIdx.x >> 5;
  const int lane = threadIdx.x & 31;
  const int n = blockIdx.x * 8 + wave;

  // binary search segment (uniform per wave)
  int lo = 0, hi = BATCH;
  while (hi - lo > 1) { int mid = (lo + hi) >> 1; if (n >= offs[mid]) lo = mid; else hi = mid; }

  const float4* mrow = (const float4*)(M + (size_t)n * DIM);
  const float4* qrow = (const float4*)(q + (size_t)lo * DIM);
  float s = 0.f;
#pragma unroll
  for (int i = 0; i < DIM / 4 / 32; ++i) {  // 8 iterations of coalesced float4
    float4 a = mrow[lane + i * 32];
    float4 b = qrow[lane + i * 32];
    s += a.x * b.x + a.y * b.y + a.z * b.z + a.w * b.w;
  }
#pragma unroll
  for (int off = 16; off > 0; off >>= 1) s += __shfl_down(s, off, 32);
  if (lane == 0) scores[n] = s;
}

// ---------------------------------------------------------------------------
// Kernel 4: per-segment stable-softmax stats (max + sum of exp)
// ---------------------------------------------------------------------------
__global__ __launch_bounds__(256) void seg_reduce_kernel(
    const float* __restrict__ scores, const int* __restrict__ offs,
    float* __restrict__ m_out, float* __restrict__ z_out) {
  __shared__ float red[256];
  const int b = blockIdx.x;
  const int s0 = offs[b], s1 = offs[b + 1];

  float mx = -__builtin_inff();
  for (int i = s0 + threadIdx.x; i < s1; i += 256) mx = fmaxf(mx, scores[i]);
  red[threadIdx.x] = mx; __syncthreads();
  for (int st = 128; st > 0; st >>= 1) {
    if (threadIdx.x < st) red[threadIdx.x] = fmaxf(red[threadIdx.x], red[threadIdx.x + st]);
    __syncthreads();
  }
  mx = red[0]; __syncthreads();

  float z = 0.f;
  for (int i = s0 + threadIdx.x; i < s1; i += 256) z += expf(scores[i] - mx);
  red[threadIdx.x] = z; __syncthreads();
  for (int st = 128; st > 0; st >>= 1) {
    if (threadIdx.x < st) red[threadIdx.x] += red[threadIdx.x + st];
    __syncthreads();
  }
  if (threadIdx.x == 0) { m_out[b] = mx; z_out[b] = red[0]; }
}

// ---------------------------------------------------------------------------
// Kernel 5: attn[n] = exp(scores[n] - m[seg]) / z[seg]
// ---------------------------------------------------------------------------
__global__ __launch_bounds__(256) void attn_kernel(
    const float* __restrict__ scores, const int* __restrict__ offs,
    const float* __restrict__ m, const float* __restrict__ z,
    float* __restrict__ attn) {
  const int n = blockIdx.x * 256 + threadIdx.x;
  int lo = 0, hi = BATCH;
  while (hi - lo > 1) { int mid = (lo + hi) >> 1; if (n >= offs[mid]) lo = mid; else hi = mid; }
  attn[n] = expf(scores[n] - m[lo]) / z[lo];
}

// ---------------------------------------------------------------------------
// Kernel 6: zero the output (it is atomically accumulated and starts poisoned)
// ---------------------------------------------------------------------------
__global__ __launch_bounds__(256) void zero_kernel(float* __restrict__ p, int n) {
  const int i = blockIdx.x * 256 + threadIdx.x;
  if (i < n) p[i] = 0.f;
}

// ---------------------------------------------------------------------------
// Kernel 7: context[b,:] += attn[n] * memory_bank[n,:]  for n in segment b.
// 128-node chunks; each of 256 threads owns 4 columns (full 1024-col row per
// node, coalesced float4). Register accumulation, atomic flush at seg bounds.
// ---------------------------------------------------------------------------
__global__ __launch_bounds__(256) void context_kernel(
    const float* __restrict__ M, const float* __restrict__ attn,
    const int* __restrict__ offs, float* __restrict__ ctx) {
  const int node0 = blockIdx.x * 128;
  const int c = threadIdx.x * 4;

  int lo = 0, hi = BATCH;
  while (hi - lo > 1) { int mid = (lo + hi) >> 1; if (node0 >= offs[mid]) lo = mid; else hi = mid; }
  int seg = lo;
  int seg_end = offs[seg + 1];

  float4 acc = make_float4(0.f, 0.f, 0.f, 0.f);
  for (int i = 0; i < 128; ++i) {
    const int n = node0 + i;
    while (n >= seg_end) {  // uniform across block
      float* dst = ctx + (size_t)seg * DIM + c;
      atomicAdd(dst + 0, acc.x); atomicAdd(dst + 1, acc.y);
      atomicAdd(dst + 2, acc.z); atomicAdd(dst + 3, acc.w);
      acc = make_float4(0.f, 0.f, 0.f, 0.f);
      ++seg;
      seg_end = offs[seg + 1];
    }
    const float w = attn[n];
    const float4 v = *(const float4*)(M + (size_t)n * DIM + c);
    acc.x += w * v.x; acc.y += w * v.y; acc.z += w * v.z; acc.w += w * v.w;
  }
  float* dst = ctx + (size_t)seg * DIM + c;
  atomicAdd(dst + 0, acc.x); atomicAdd(dst + 1, acc.y);
  atomicAdd(dst + 2, acc.z); atomicAdd(dst + 3, acc.w);
}

// ---------------------------------------------------------------------------
extern "C" void kernel_launch(void* const* d_in, const int* in_sizes, int n_in,
                              void* d_out, int out_size, void* d_ws, size_t ws_size,
                              hipStream_t stream) {
  const float*     memory_bank   = (const float*)d_in[0];      // [N, 1024]
  const float*     decoder_state = (const float*)d_in[1];      // [64, 1024]
  const float*     W             = (const float*)d_in[2];      // [1024, 1024]
  const long long* lens          = (const long long*)d_in[3];  // [64] int64
  float* ctx = (float*)d_out;                                  // [64, 1024]

  // Workspace layout (all sizes padded; total < 1 MB)
  char*  ws     = (char*)d_ws;
  int*   offs   = (int*)ws;                          // 65 ints
  float* q      = (float*)(ws + 512);                // 64*1024 f32  = 256 KB
  float* scores = (float*)(ws + 512 + 262144);       // 65536 f32    = 256 KB
  float* attn   = (float*)(ws + 512 + 2 * 262144);   // 65536 f32    = 256 KB
  float* mbuf   = (float*)(ws + 512 + 3 * 262144);   // 64 f32
  float* zbuf   = (float*)(ws + 512 + 3 * 262144 + 256);  // 64 f32

  seg_offsets_kernel<<<dim3(1), dim3(32), 0, stream>>>(lens, offs);
  qgemm_wmma_kernel<<<dim3(DIM / 16, BATCH / 16), dim3(32), 0, stream>>>(
      decoder_state, W, q);
  scores_kernel<<<dim3(N_NODES / 8), dim3(256), 0, stream>>>(
      memory_bank, q, offs, scores);
  seg_reduce_kernel<<<dim3(BATCH), dim3(256), 0, stream>>>(scores, offs, mbuf, zbuf);
  attn_kernel<<<dim3(N_NODES / 256), dim3(256), 0, stream>>>(scores, offs, mbuf, zbuf, attn);
  zero_kernel<<<dim3((BATCH * DIM + 255) / 256), dim3(256), 0, stream>>>(ctx, BATCH * DIM);
  context_kernel<<<dim3(N_NODES / 128), dim3(256), 0, stream>>>(
      memory_bank, attn, offs, ctx);
}